// CoeffPredLSTM_12171937317406
// MI455X (gfx1250) — compile-verified
//
#include <hip/hip_runtime.h>
#include <hip/hip_bf16.h>
#include <math.h>

// ---------------------------------------------------------------------------
// Seq2seq LSTM + attention for MI455X (gfx1250, wave32, WMMA).
// fp32 end-to-end via V_WMMA_F32_16X16X4_F32 (256-step recurrences; GEMM
// stages are HBM/L2-bound so bf16 WMMA density wouldn't help end-to-end).
// B matrices are pre-packed so each lane's (k,k+1) pair is one aligned 8-byte
// word -> single global_load_b64 per fragment, no exec-mask predication in
// any WMMA loop. __builtin_prefetch streams ahead (global_prefetch_b8).
// ---------------------------------------------------------------------------

#define BB 64
#define SS 256
#define TT 256   // decoder steps = TGT-1
#define DD 256
#define HH 256
#define GG 1024
#define VV 1000

typedef __attribute__((ext_vector_type(2))) float v2f;
typedef __attribute__((ext_vector_type(8))) float v8f;

__device__ __forceinline__ v8f wmma4(v2f a, v2f b, v8f c) {
  // D = A(16x4 f32) x B(4x16 f32) + C(16x16 f32)
  return __builtin_amdgcn_wmma_f32_16x16x4_f32(false, a, false, b, (short)0, c,
                                               false, false);
}

__device__ __forceinline__ float sigf(float x) {
  return 1.0f / (1.0f + __expf(-x));
}

// ---------------------------------------------------------------------------
// Weight pack: W (N,K) row-major -> WP pair-packed:
//   WP[(k/2)*(2N) + 2n + (k&1)] = W[n*K + k]
// so a lane's B fragment {W[n][kk], W[n][kk+1]} is one aligned 8-byte load.
// ---------------------------------------------------------------------------
__global__ void k_pack(const float* __restrict__ W, float* __restrict__ WP,
                       int N, int K) {
  int i = blockIdx.x * blockDim.x + threadIdx.x;
  if (i >= N * K) return;
  int n = i / K, k = i - n * K;
  WP[(size_t)(k >> 1) * (2 * N) + 2 * n + (k & 1)] = W[i];
}

// ---------------------------------------------------------------------------
// Embedding gather. Row r of `out` = emb[ids[b*id_row_len + t]], r = b*id_cols+t.
// ---------------------------------------------------------------------------
__global__ void k_embed(const int* __restrict__ ids, const float* __restrict__ emb,
                        float* __restrict__ out, int rows, int id_row_len,
                        int id_cols) {
  int i = blockIdx.x * blockDim.x + threadIdx.x;
  if (i >= rows * DD) return;
  int r = i >> 8, d = i & (DD - 1);
  int b = r / id_cols, t = r - b * id_cols;
  int id = ids[b * id_row_len + t];
  out[i] = emb[id * DD + d];
}

// ---------------------------------------------------------------------------
// Generic WMMA GEMM: C[M,N] = act(A[M,K](lda) @ W^T + bias1 + bias2), with
// BP = pair-packed W^T (pitch 2N). M multiple of 16; N tail handled by
// clamped loads (garbage columns are never stored). act: 0 none, 1 tanh.
// ---------------------------------------------------------------------------
__global__ __launch_bounds__(256) void k_gemm_wmma(
    const float* __restrict__ A, int lda, const float* __restrict__ BP,
    float* __restrict__ C, int ldc, int M, int N, int K,
    const float* __restrict__ bias1, const float* __restrict__ bias2, int act) {
  const int lane = threadIdx.x & 31;
  const int wave = threadIdx.x >> 5;
  const int m0 = blockIdx.x * 16;
  const int n0 = blockIdx.y * 128 + wave * 16;
  if (n0 >= N) return;  // wave-uniform: EXEC stays all-ones for WMMA waves
  const int ncol = n0 + (lane & 15);
  const int ncl = (ncol < N) ? ncol : (N - 1);  // clamp instead of predicate
  const int khalf = (lane >> 4) << 1;           // lanes 16..31: K+2,K+3

  const float* ap = A + (size_t)(m0 + (lane & 15)) * lda + khalf;
  const float* bp = BP + (size_t)(khalf >> 1) * (2 * N) + 2 * ncl;
  const size_t bstep = (size_t)4 * N;  // advance 2 pairs per k+=4

  v8f acc = {};
#pragma unroll 2
  for (int k = 0; k < K; k += 4) {
    __builtin_prefetch(bp + 16 * bstep, 0, 0);  // stream B panel ahead
    v2f a = *(const v2f*)ap;
    v2f b = *(const v2f*)bp;
    acc = wmma4(a, b, acc);
    ap += 4;
    bp += bstep;
  }
  if (ncol >= N) return;
  float bs = 0.0f;
  if (bias1) bs += bias1[ncol];
  if (bias2) bs += bias2[ncol];
#pragma unroll
  for (int r = 0; r < 8; ++r) {
    const int row = m0 + r + ((lane >> 4) << 3);
    float v = acc[r] + bs;
    if (act == 1) v = tanhf(v);
    C[(size_t)row * ldc + ncol] = v;
  }
}

// ---------------------------------------------------------------------------
// LSTM scan. Grid = 4 blocks (batch tiles of 16), block = 512 threads (16
// waves). Wave w owns hidden slice j in [16w,16w+16) and computes the four
// gate tiles (i,f,g,o) with WMMA; c-state lives in registers for all steps.
// pre[(b*S+t)*G + n] already holds x@Wih^T + bih + bhh. WhhP = pair-packed
// Whh^T (pitch 2G).
// ---------------------------------------------------------------------------
__global__ __launch_bounds__(512) void k_lstm_scan(
    const float* __restrict__ pre, const float* __restrict__ WhhP,
    float* __restrict__ ys, int ys_ld, int ys_off,
    float* __restrict__ hT, float* __restrict__ cT, int st_ld, int st_off,
    int reverse) {
  __shared__ float hbuf[16 * HH];
  const int lane = threadIdx.x & 31;
  const int wave = threadIdx.x >> 5;
  const int bt = blockIdx.x * 16;
  const int j = wave * 16 + (lane & 15);
  const int khalf = (lane >> 4) << 1;
  const int hi8 = (lane >> 4) << 3;
  const int mrow = lane & 15;

  for (int i = threadIdx.x; i < 16 * HH; i += 512) hbuf[i] = 0.0f;
  v8f creg = {};
  __syncthreads();

  for (int tt = 0; tt < SS; ++tt) {
    const int t = reverse ? (SS - 1 - tt) : tt;
    // prefetch next step's pre row for this lane's slice
    const int tn = reverse ? (t - 1) : (t + 1);
    if (tn >= 0 && tn < SS)
      __builtin_prefetch(pre + ((size_t)(bt + hi8) * SS + tn) * GG + j, 0, 0);

    v8f a0 = {}, a1 = {}, a2 = {}, a3 = {};
    const float* hp = hbuf + mrow * HH + khalf;
    const float* bp = WhhP + (size_t)(khalf >> 1) * (2 * GG);
    const size_t bstep = (size_t)4 * GG;
#pragma unroll 2
    for (int k = 0; k < HH; k += 4) {
      v2f av = *(const v2f*)hp;
      v2f b;
      b = *(const v2f*)(bp + 2 * j);              a0 = wmma4(av, b, a0);
      b = *(const v2f*)(bp + 2 * (HH + j));       a1 = wmma4(av, b, a1);
      b = *(const v2f*)(bp + 2 * (2 * HH + j));   a2 = wmma4(av, b, a2);
      b = *(const v2f*)(bp + 2 * (3 * HH + j));   a3 = wmma4(av, b, a3);
      hp += 4;
      bp += bstep;
    }
    float hst[8];
#pragma unroll
    for (int r = 0; r < 8; ++r) {
      const int br = r + hi8;
      const size_t prow = (size_t)(bt + br) * SS + t;
      const float* pr = pre + prow * GG;
      const float gi = a0[r] + pr[j];
      const float gf = a1[r] + pr[HH + j];
      const float gg = a2[r] + pr[2 * HH + j];
      const float go = a3[r] + pr[3 * HH + j];
      const float c = sigf(gf) * creg[r] + sigf(gi) * tanhf(gg);
      creg[r] = c;
      const float h = sigf(go) * tanhf(c);
      hst[r] = h;
      ys[prow * ys_ld + ys_off + j] = h;
      if (tt == SS - 1) {
        if (hT) hT[(bt + br) * st_ld + st_off + j] = h;
        if (cT) cT[(bt + br) * st_ld + st_off + j] = c;
      }
    }
    __syncthreads();  // all waves done reading hbuf
#pragma unroll
    for (int r = 0; r < 8; ++r) hbuf[(r + hi8) * HH + j] = hst[r];
    __syncthreads();
  }
}

// ---------------------------------------------------------------------------
// Decoder scan: attention + 2-layer LSTM, 256 steps. Grid = 4 batch tiles,
// 512 threads. q-buffer aliases the first 16KB of the ctx buffer (q is dead
// by the time ctx is written). All weight operands pair-packed (pitch 2G/2H).
// ---------------------------------------------------------------------------
__global__ __launch_bounds__(512) void k_decoder_scan(
    const float* __restrict__ pre_dec,     // (B*T, G): xt@Wih_x^T + biases
    const float* __restrict__ d0_WihP_ctx, // packed rows 256..767, pitch 2G
    const float* __restrict__ d0_WhhP,     // packed (H,G)
    const float* __restrict__ d1_WihP,     // packed (H,G)
    const float* __restrict__ d1_WhhP,     // packed (H,G)
    const float* __restrict__ d1_bih, const float* __restrict__ d1_bhh,
    const float* __restrict__ h0, const float* __restrict__ c0,  // (B, H)
    const float* __restrict__ enc_out,    // (B*S, 2H)
    const float* __restrict__ enc_proj,   // (B*S, H)
    const float* __restrict__ att_WdecP,  // packed (H,H), pitch 2H
    const float* __restrict__ att_v,      // (H)
    const int* __restrict__ src,          // (B, S)
    float* __restrict__ outh)             // (B*T, H)
{
  __shared__ float hl0[16 * HH];
  __shared__ float hl1[16 * HH];
  __shared__ float wsc[16 * SS];
  __shared__ float ctxq[16 * 2 * HH];  // ctx (16,512); first 16*HH doubles as q
  __shared__ float avs[HH];

  const int lane = threadIdx.x & 31;
  const int wave = threadIdx.x >> 5;
  const int bt = blockIdx.x * 16;
  const int j = wave * 16 + (lane & 15);
  const int khalf = (lane >> 4) << 1;
  const int hi8 = (lane >> 4) << 3;
  const int mrow = lane & 15;

  for (int i = threadIdx.x; i < 16 * HH; i += 512) {
    const int m = i >> 8, jj = i & (HH - 1);
    const float v = h0[(bt + m) * HH + jj];
    hl0[i] = v;
    hl1[i] = v;
  }
  if (threadIdx.x < HH) avs[threadIdx.x] = att_v[threadIdx.x];
  v8f c0reg, c1reg;
#pragma unroll
  for (int r = 0; r < 8; ++r) {
    const float cv = c0[(bt + r + hi8) * HH + j];
    c0reg[r] = cv;
    c1reg[r] = cv;
  }
  const float b1g0 = d1_bih[j] + d1_bhh[j];
  const float b1g1 = d1_bih[HH + j] + d1_bhh[HH + j];
  const float b1g2 = d1_bih[2 * HH + j] + d1_bhh[2 * HH + j];
  const float b1g3 = d1_bih[3 * HH + j] + d1_bhh[3 * HH + j];
  __syncthreads();

  for (int t = 0; t < TT; ++t) {
    // ---- q = hl1 @ att_Wdec^T : 16 waves x one 16x16 tile
    {
      v8f q = {};
      const float* hp = hl1 + mrow * HH + khalf;
      const float* bp = att_WdecP + (size_t)(khalf >> 1) * (2 * HH) + 2 * j;
#pragma unroll 2
      for (int k = 0; k < HH; k += 4) {
        v2f av = *(const v2f*)hp;
        v2f b = *(const v2f*)bp;
        q = wmma4(av, b, q);
        hp += 4;
        bp += 4 * HH;
      }
#pragma unroll
      for (int r = 0; r < 8; ++r) ctxq[(r + hi8) * HH + j] = q[r];
    }
    __syncthreads();
    // ---- scores[b][s] = sum_h tanh(enc_proj + q) * att_v, then mask
    for (int p = threadIdx.x; p < 16 * SS; p += 512) {
      const int b = p >> 8, s = p & (SS - 1);
      const float* ep = enc_proj + ((size_t)(bt + b) * SS + s) * HH;
      const float* qb = ctxq + b * HH;
      float sc = 0.0f;
      for (int h = 0; h < HH; ++h) sc += tanhf(ep[h] + qb[h]) * avs[h];
      if (src[(bt + b) * SS + s] == 0) sc = -INFINITY;
      wsc[p] = sc;
    }
    __syncthreads();
    // ---- softmax over s per batch row
    if (threadIdx.x < 16) {
      const int b = threadIdx.x;
      float mx = -INFINITY;
      for (int s = 0; s < SS; ++s) mx = fmaxf(mx, wsc[b * SS + s]);
      float sum = 0.0f;
      for (int s = 0; s < SS; ++s) {
        const float e = __expf(wsc[b * SS + s] - mx);
        wsc[b * SS + s] = e;
        sum += e;
      }
      const float inv = 1.0f / sum;
      for (int s = 0; s < SS; ++s) wsc[b * SS + s] *= inv;
    }
    __syncthreads();
    // ---- ctx[b][e] = sum_s w[b][s] * enc_out[b][s][e]
    {
      const int b = threadIdx.x >> 5;
      const int e0 = (threadIdx.x & 31) * 16;
      float acc[16];
#pragma unroll
      for (int e = 0; e < 16; ++e) acc[e] = 0.0f;
      for (int s = 0; s < SS; ++s) {
        const float wv = wsc[b * SS + s];
        const float* eo = enc_out + ((size_t)(bt + b) * SS + s) * (2 * HH) + e0;
#pragma unroll
        for (int e = 0; e < 16; ++e) acc[e] += wv * eo[e];
      }
#pragma unroll
      for (int e = 0; e < 16; ++e) ctxq[b * (2 * HH) + e0 + e] = acc[e];
    }
    __syncthreads();
    // ---- layer-0 gates: ctx@Wih_ctx^T + hl0@Whh^T + pre_dec, then cell
    {
      v8f g0 = {}, g1 = {}, g2 = {}, g3 = {};
      {
        const float* hp = ctxq + mrow * (2 * HH) + khalf;
        const float* bp = d0_WihP_ctx + (size_t)(khalf >> 1) * (2 * GG);
#pragma unroll 2
        for (int k = 0; k < 2 * HH; k += 4) {
          v2f av = *(const v2f*)hp;
          v2f b;
          b = *(const v2f*)(bp + 2 * j);            g0 = wmma4(av, b, g0);
          b = *(const v2f*)(bp + 2 * (HH + j));     g1 = wmma4(av, b, g1);
          b = *(const v2f*)(bp + 2 * (2 * HH + j)); g2 = wmma4(av, b, g2);
          b = *(const v2f*)(bp + 2 * (3 * HH + j)); g3 = wmma4(av, b, g3);
          hp += 4;
          bp += (size_t)4 * GG;
        }
      }
      {
        const float* hp = hl0 + mrow * HH + khalf;
        const float* bp = d0_WhhP + (size_t)(khalf >> 1) * (2 * GG);
#pragma unroll 2
        for (int k = 0; k < HH; k += 4) {
          v2f av = *(const v2f*)hp;
          v2f b;
          b = *(const v2f*)(bp + 2 * j);            g0 = wmma4(av, b, g0);
          b = *(const v2f*)(bp + 2 * (HH + j));     g1 = wmma4(av, b, g1);
          b = *(const v2f*)(bp + 2 * (2 * HH + j)); g2 = wmma4(av, b, g2);
          b = *(const v2f*)(bp + 2 * (3 * HH + j)); g3 = wmma4(av, b, g3);
          hp += 4;
          bp += (size_t)4 * GG;
        }
      }
      float hst[8];
#pragma unroll
      for (int r = 0; r < 8; ++r) {
        const int br = r + hi8;
        const float* pr = pre_dec + ((size_t)(bt + br) * TT + t) * GG;
        const float gi = g0[r] + pr[j];
        const float gf = g1[r] + pr[HH + j];
        const float gg = g2[r] + pr[2 * HH + j];
        const float go = g3[r] + pr[3 * HH + j];
        const float c = sigf(gf) * c0reg[r] + sigf(gi) * tanhf(gg);
        c0reg[r] = c;
        hst[r] = sigf(go) * tanhf(c);
      }
      __syncthreads();  // all waves done reading hl0 / ctxq
#pragma unroll
      for (int r = 0; r < 8; ++r) hl0[(r + hi8) * HH + j] = hst[r];
    }
    __syncthreads();
    // ---- layer-1 gates: hl0@d1_Wih^T + hl1@d1_Whh^T + biases, then cell
    {
      v8f g0 = {}, g1 = {}, g2 = {}, g3 = {};
      {
        const float* hp = hl0 + mrow * HH + khalf;
        const float* bp = d1_WihP + (size_t)(khalf >> 1) * (2 * GG);
#pragma unroll 2
        for (int k = 0; k < HH; k += 4) {
          v2f av = *(const v2f*)hp;
          v2f b;
          b = *(const v2f*)(bp + 2 * j);            g0 = wmma4(av, b, g0);
          b = *(const v2f*)(bp + 2 * (HH + j));     g1 = wmma4(av, b, g1);
          b = *(const v2f*)(bp + 2 * (2 * HH + j)); g2 = wmma4(av, b, g2);
          b = *(const v2f*)(bp + 2 * (3 * HH + j)); g3 = wmma4(av, b, g3);
          hp += 4;
          bp += (size_t)4 * GG;
        }
      }
      {
        const float* hp = hl1 + mrow * HH + khalf;
        const float* bp = d1_WhhP + (size_t)(khalf >> 1) * (2 * GG);
#pragma unroll 2
        for (int k = 0; k < HH; k += 4) {
          v2f av = *(const v2f*)hp;
          v2f b;
          b = *(const v2f*)(bp + 2 * j);            g0 = wmma4(av, b, g0);
          b = *(const v2f*)(bp + 2 * (HH + j));     g1 = wmma4(av, b, g1);
          b = *(const v2f*)(bp + 2 * (2 * HH + j)); g2 = wmma4(av, b, g2);
          b = *(const v2f*)(bp + 2 * (3 * HH + j)); g3 = wmma4(av, b, g3);
          hp += 4;
          bp += (size_t)4 * GG;
        }
      }
      float hst[8];
#pragma unroll
      for (int r = 0; r < 8; ++r) {
        const int br = r + hi8;
        const float gi = g0[r] + b1g0;
        const float gf = g1[r] + b1g1;
        const float gg = g2[r] + b1g2;
        const float go = g3[r] + b1g3;
        const float c = sigf(gf) * c1reg[r] + sigf(gi) * tanhf(gg);
        c1reg[r] = c;
        const float h = sigf(go) * tanhf(c);
        hst[r] = h;
        outh[((size_t)(bt + br) * TT + t) * HH + j] = h;
      }
      __syncthreads();  // all waves done reading hl1
#pragma unroll
      for (int r = 0; r < 8; ++r) hl1[(r + hi8) * HH + j] = hst[r];
    }
    __syncthreads();
  }
}

// ---------------------------------------------------------------------------
// Host orchestration.
// ---------------------------------------------------------------------------
extern "C" void kernel_launch(void* const* d_in, const int* in_sizes, int n_in,
                              void* d_out, int out_size, void* d_ws,
                              size_t ws_size, hipStream_t stream) {
  (void)in_sizes; (void)n_in; (void)out_size; (void)ws_size;
  const int* src = (const int*)d_in[0];
  const int* tgt = (const int*)d_in[1];
  const float* emb = (const float*)d_in[2];
  const float* e0f_Wih = (const float*)d_in[3];
  const float* e0f_Whh = (const float*)d_in[4];
  const float* e0f_bih = (const float*)d_in[5];
  const float* e0f_bhh = (const float*)d_in[6];
  const float* e0b_Wih = (const float*)d_in[7];
  const float* e0b_Whh = (const float*)d_in[8];
  const float* e0b_bih = (const float*)d_in[9];
  const float* e0b_bhh = (const float*)d_in[10];
  const float* e1f_Wih = (const float*)d_in[11];
  const float* e1f_Whh = (const float*)d_in[12];
  const float* e1f_bih = (const float*)d_in[13];
  const float* e1f_bhh = (const float*)d_in[14];
  const float* e1b_Wih = (const float*)d_in[15];
  const float* e1b_Whh = (const float*)d_in[16];
  const float* e1b_bih = (const float*)d_in[17];
  const float* e1b_bhh = (const float*)d_in[18];
  const float* d0_Wih = (const float*)d_in[19];
  const float* d0_Whh = (const float*)d_in[20];
  const float* d0_bih = (const float*)d_in[21];
  const float* d0_bhh = (const float*)d_in[22];
  const float* d1_Wih = (const float*)d_in[23];
  const float* d1_Whh = (const float*)d_in[24];
  const float* d1_bih = (const float*)d_in[25];
  const float* d1_bhh = (const float*)d_in[26];
  const float* h_proj_W = (const float*)d_in[27];
  const float* h_proj_b = (const float*)d_in[28];
  const float* c_proj_W = (const float*)d_in[29];
  const float* c_proj_b = (const float*)d_in[30];
  const float* att_Wenc = (const float*)d_in[31];
  const float* att_Wdec = (const float*)d_in[32];
  const float* att_v = (const float*)d_in[33];
  float* logits = (float*)d_out;

  float* ws = (float*)d_ws;
  size_t off = 0;
  auto alloc = [&](size_t n) { float* p = ws + off; off += n; return p; };

  // Pair-packed weights (persistent).
  float* e0f_WihP = alloc(DD * GG);
  float* e0b_WihP = alloc(DD * GG);
  float* e1f_WihP = alloc(2 * HH * GG);
  float* e1b_WihP = alloc(2 * HH * GG);
  float* e0f_WhhP = alloc(HH * GG);
  float* e0b_WhhP = alloc(HH * GG);
  float* e1f_WhhP = alloc(HH * GG);
  float* e1b_WhhP = alloc(HH * GG);
  float* d0_WihP = alloc((DD + 2 * HH) * GG);
  float* d0_WhhP = alloc(HH * GG);
  float* d1_WihP = alloc(HH * GG);
  float* d1_WhhP = alloc(HH * GG);
  float* h_projP = alloc(2 * HH * HH);
  float* c_projP = alloc(2 * HH * HH);
  float* att_WencP = alloc(2 * HH * HH);
  float* att_WdecP = alloc(HH * HH);
  float* embP = alloc(DD * VV);

  // Activations (with reuse).
  float* x0 = alloc((size_t)BB * SS * DD);    // also tgt_emb later
  float* x1 = alloc((size_t)BB * SS * 2 * HH);
  float* preA = alloc((size_t)BB * SS * GG);  // pre_e0f/e1f, later pre_dec
  float* preB = alloc((size_t)BB * SS * GG);  // pre_e0b/e1b, later outh
  float* enc_out = alloc((size_t)BB * SS * 2 * HH);
  float* enc_proj = alloc((size_t)BB * SS * HH);
  float* hcat = alloc(BB * 2 * HH);
  float* ccat = alloc(BB * 2 * HH);
  float* h0 = alloc(BB * HH);
  float* c0 = alloc(BB * HH);

  auto pk = [&](const float* W, float* WP, int N, int K) {
    const int total = N * K;
    k_pack<<<(total + 255) / 256, 256, 0, stream>>>(W, WP, N, K);
  };
  auto gemm = [&](const float* A, int lda, const float* BP, float* C, int ldc,
                  int M, int N, int K, const float* b1, const float* b2,
                  int act) {
    dim3 grid(M / 16, (N + 127) / 128);
    k_gemm_wmma<<<grid, 256, 0, stream>>>(A, lda, BP, C, ldc, M, N, K, b1, b2,
                                          act);
  };

  // 1) Pack all weights.
  pk(e0f_Wih, e0f_WihP, GG, DD);
  pk(e0b_Wih, e0b_WihP, GG, DD);
  pk(e1f_Wih, e1f_WihP, GG, 2 * HH);
  pk(e1b_Wih, e1b_WihP, GG, 2 * HH);
  pk(e0f_Whh, e0f_WhhP, GG, HH);
  pk(e0b_Whh, e0b_WhhP, GG, HH);
  pk(e1f_Whh, e1f_WhhP, GG, HH);
  pk(e1b_Whh, e1b_WhhP, GG, HH);
  pk(d0_Wih, d0_WihP, GG, DD + 2 * HH);
  pk(d0_Whh, d0_WhhP, GG, HH);
  pk(d1_Wih, d1_WihP, GG, HH);
  pk(d1_Whh, d1_WhhP, GG, HH);
  pk(h_proj_W, h_projP, HH, 2 * HH);
  pk(c_proj_W, c_projP, HH, 2 * HH);
  pk(att_Wenc, att_WencP, HH, 2 * HH);
  pk(att_Wdec, att_WdecP, HH, HH);
  pk(emb, embP, VV, DD);

  const int MR = BB * SS;  // 16384 rows

  // 2) Embed src -> x0.
  k_embed<<<(MR * DD + 255) / 256, 256, 0, stream>>>(src, emb, x0, MR, SS, SS);

  // 3) Layer-0 input projections (+ both biases folded in).
  gemm(x0, DD, e0f_WihP, preA, GG, MR, GG, DD, e0f_bih, e0f_bhh, 0);
  gemm(x0, DD, e0b_WihP, preB, GG, MR, GG, DD, e0b_bih, e0b_bhh, 0);

  // 4) Layer-0 scans -> x1 = [f0 | b0].
  k_lstm_scan<<<4, 512, 0, stream>>>(preA, e0f_WhhP, x1, 2 * HH, 0,
                                     (float*)nullptr, (float*)nullptr, 0, 0, 0);
  k_lstm_scan<<<4, 512, 0, stream>>>(preB, e0b_WhhP, x1, 2 * HH, HH,
                                     (float*)nullptr, (float*)nullptr, 0, 0, 1);

  // 5) Layer-1 input projections.
  gemm(x1, 2 * HH, e1f_WihP, preA, GG, MR, GG, 2 * HH, e1f_bih, e1f_bhh, 0);
  gemm(x1, 2 * HH, e1b_WihP, preB, GG, MR, GG, 2 * HH, e1b_bih, e1b_bhh, 0);

  // 6) Layer-1 scans -> enc_out = [f1 | b1]; final states -> hcat/ccat.
  k_lstm_scan<<<4, 512, 0, stream>>>(preA, e1f_WhhP, enc_out, 2 * HH, 0, hcat,
                                     ccat, 2 * HH, 0, 0);
  k_lstm_scan<<<4, 512, 0, stream>>>(preB, e1b_WhhP, enc_out, 2 * HH, HH, hcat,
                                     ccat, 2 * HH, HH, 1);

  // 7) Initial decoder states: h0 = tanh(hcat @ h_proj^T + b), c0 likewise.
  gemm(hcat, 2 * HH, h_projP, h0, HH, BB, HH, 2 * HH, h_proj_b,
       (const float*)nullptr, 1);
  gemm(ccat, 2 * HH, c_projP, c0, HH, BB, HH, 2 * HH, c_proj_b,
       (const float*)nullptr, 1);

  // 8) enc_proj = enc_out @ att_Wenc^T.
  gemm(enc_out, 2 * HH, att_WencP, enc_proj, HH, MR, HH, 2 * HH,
       (const float*)nullptr, (const float*)nullptr, 0);

  // 9) Embed tgt[:, :-1] -> x0 (reused as tgt_emb).
  k_embed<<<(BB * TT * DD + 255) / 256, 256, 0, stream>>>(tgt, emb, x0, BB * TT,
                                                          SS + 1, TT);

  // 10) Decoder input projection (xt part only, K=256) + biases -> preA.
  gemm(x0, DD, d0_WihP, preA, GG, BB * TT, GG, DD, d0_bih, d0_bhh, 0);

  // 11) Decoder scan -> preB reused as hidden outputs (B*T, H).
  k_decoder_scan<<<4, 512, 0, stream>>>(
      preA, d0_WihP + (size_t)DD * GG, d0_WhhP, d1_WihP, d1_WhhP, d1_bih,
      d1_bhh, h0, c0, enc_out, enc_proj, att_WdecP, att_v, src, preB);

  // 12) Logits = outh @ emb^T -> d_out (B*T, VOCAB).
  gemm(preB, HH, embP, logits, VV, BB * TT, VV, DD, (const float*)nullptr,
       (const float*)nullptr, 0);
}